// NSLayer_26886495273251
// MI455X (gfx1250) — compile-verified
//
#include <hip/hip_runtime.h>
#include <hip/hip_bf16.h>

typedef __attribute__((ext_vector_type(2))) float v2f;
typedef __attribute__((ext_vector_type(4))) float v4f;
typedef __attribute__((ext_vector_type(8))) float v8f;

// Padded LDS row stride (floats). 20 => 80B row: 16B-aligned float4 staging,
// and m*20 mod 64 distinct for m=0..15 => conflict-free A-frag b64 loads.
#define LROW 20
#define WAVES_PER_BLOCK 8
#define N_TERMS 14

// D = A*B + C as 16x16x16 f32 GEMM using 4 chained V_WMMA_F32_16X16X4_F32.
// a[g] : A-fragment for K-group g (lane<16: rows, K=4g+2hi+{0,1})
// b[g] : B-fragment for K-group g
__device__ __forceinline__ v8f gemm16(const v2f a[4], const v2f b[4], v8f c) {
#pragma unroll
    for (int g = 0; g < 4; ++g) {
        c = __builtin_amdgcn_wmma_f32_16x16x4_f32(
                /*neg_a=*/false, a[g], /*neg_b=*/false, b[g],
                /*c_mod=*/(short)0, c, /*reuse_a=*/false, /*reuse_b=*/false);
    }
    return c;
}

__device__ __forceinline__ void lds_wait() {
    asm volatile("s_wait_dscnt 0" ::: "memory");
}

__global__ __launch_bounds__(WAVES_PER_BLOCK * 32)
void ns_wmma_kernel(const float* __restrict__ X,
                    const float* __restrict__ W,
                    float* __restrict__ out,
                    int nMat) {
    // Two private 16x(LROW) f32 tiles per wave: [0] = X tile, [1] = scratch.
    __shared__ __align__(16) float lds[WAVES_PER_BLOCK][2][16 * LROW];

    const int lane = threadIdx.x & 31;
    const int wave = threadIdx.x >> 5;
    const long long mat = (long long)blockIdx.x * WAVES_PER_BLOCK + wave;
    if (mat >= nMat) return;                 // wave-uniform exit (EXEC all-0 or all-1)

    const float* __restrict__ src = X + mat * 256;
    float* __restrict__ dst = out + mat * 256;
    float* xt = &lds[wave][0][0];            // X tile (kept for final Mat@X + X)
    float* st = &lds[wave][1][0];            // scratch tile for layout conversion

    const int hi = lane >> 4;                // lane half: 0 or 1
    const int m  = lane & 15;                // row/col index within half

    // ---- coalesced load of X into padded LDS (2x global_load_b128 per lane) ----
    {
        const int r = lane >> 1, cb = (lane & 1) * 8;
        const v4f* g = (const v4f*)(src + lane * 8);
        v4f p0 = g[0];
        v4f p1 = g[1];
        *(v4f*)(xt + r * LROW + cb)     = p0;
        *(v4f*)(xt + r * LROW + cb + 4) = p1;
    }
    lds_wait();

    // ---- S = X * X^T : B-frag(X^T) == A-frag(X), so one fragment feeds both ----
    v8f acc = {};
    {
        v2f ax[4];
#pragma unroll
        for (int g = 0; g < 4; ++g)
            ax[g] = *(const v2f*)(xt + m * LROW + 4 * g + 2 * hi);
        acc = gemm16(ax, ax, acc);
    }

    // ---- A = I - S (elementwise in C/D layout: M = r + 8*hi, N = m) ----
    v8f cur, Macc;
    {
        const float w0 = __builtin_fabsf(W[0]);
#pragma unroll
        for (int r = 0; r < 8; ++r) {
            float id = ((r + 8 * hi) == m) ? 1.0f : 0.0f;
            cur[r]  = id - acc[r];
            Macc[r] = w0 * cur[r];
        }
    }

    // ---- repeated squaring: cur <- cur*cur; Macc += |w_n| * cur ----
    for (int n = 1; n < N_TERMS; ++n) {
        // C/D layout -> LDS scratch tile
#pragma unroll
        for (int r = 0; r < 8; ++r)
            st[(r + 8 * hi) * LROW + m] = cur[r];
        lds_wait();

        v2f a[4], b[4];
#pragma unroll
        for (int g = 0; g < 4; ++g) {
            a[g] = *(const v2f*)(st + m * LROW + 4 * g + 2 * hi);   // b64, conflict-free
            b[g].x = st[(4 * g + 0 + 2 * hi) * LROW + m];
            b[g].y = st[(4 * g + 1 + 2 * hi) * LROW + m];
        }
        v8f sq = {};
        sq = gemm16(a, b, sq);
        cur = sq;

        const float wn = __builtin_fabsf(W[n]);   // uniform -> scalar load
#pragma unroll
        for (int r = 0; r < 8; ++r)
            Macc[r] += wn * cur[r];
    }

    // ---- res = Macc @ X  (Macc -> A-frags via LDS; X B-frags from xt) ----
#pragma unroll
    for (int r = 0; r < 8; ++r)
        st[(r + 8 * hi) * LROW + m] = Macc[r];
    lds_wait();

    v8f res = {};
    {
        v2f a[4], b[4];
#pragma unroll
        for (int g = 0; g < 4; ++g) {
            a[g] = *(const v2f*)(st + m * LROW + 4 * g + 2 * hi);
            b[g].x = xt[(4 * g + 0 + 2 * hi) * LROW + m];
            b[g].y = xt[(4 * g + 1 + 2 * hi) * LROW + m];
        }
        res = gemm16(a, b, res);
    }

    // ---- out = X + res ----
#pragma unroll
    for (int r = 0; r < 8; ++r)
        res[r] += xt[(r + 8 * hi) * LROW + m];

    // stage through LDS for coalesced global_store_b128
#pragma unroll
    for (int r = 0; r < 8; ++r)
        st[(r + 8 * hi) * LROW + m] = res[r];
    lds_wait();
    {
        const int r = lane >> 1, cb = (lane & 1) * 8;
        v4f p0 = *(const v4f*)(st + r * LROW + cb);
        v4f p1 = *(const v4f*)(st + r * LROW + cb + 4);
        v4f* g = (v4f*)(dst + lane * 8);
        g[0] = p0;
        g[1] = p1;
    }
}

extern "C" void kernel_launch(void* const* d_in, const int* in_sizes, int n_in,
                              void* d_out, int out_size, void* d_ws, size_t ws_size,
                              hipStream_t stream) {
    const float* X = (const float*)d_in[0];   // [64,512,16,16] f32
    const float* W = (const float*)d_in[1];   // [14] f32
    float* out = (float*)d_out;

    const int nMat = in_sizes[0] / 256;       // 32768 matrices
    const int blocks = (nMat + WAVES_PER_BLOCK - 1) / WAVES_PER_BLOCK;
    ns_wmma_kernel<<<blocks, WAVES_PER_BLOCK * 32, 0, stream>>>(X, W, out, nMat);
}